// GraphDecoder_60103772340707
// MI455X (gfx1250) — compile-verified
//
#include <hip/hip_runtime.h>
#include <hip/hip_bf16.h>

typedef __attribute__((ext_vector_type(2))) float v2f;
typedef __attribute__((ext_vector_type(8))) float v8f;

#define NEG_INF (-__builtin_inff())

#define NGRAPHS 512
#define DIM     256
#define HID     512

// ---------------------------------------------------------------------------
// Init: fill pooling buffer with -inf
// ---------------------------------------------------------------------------
__global__ __launch_bounds__(256) void fill_neginf_kernel(float* __restrict__ g, int n) {
    int i = blockIdx.x * 256 + threadIdx.x;
    if (i < n) g[i] = NEG_INF;
}

// ---------------------------------------------------------------------------
// Segment max over sorted batch vector.
// Block handles 512 nodes, split into 4 sub-streams of 128 nodes.
// Within a sub-stream, 64 threads each own 4 features (float4, fully
// coalesced 1 KB row loads). Runs of equal batch-id are reduced in registers;
// atomic f32 max only at segment boundaries (lands in L2).
// ---------------------------------------------------------------------------
__global__ __launch_bounds__(256) void seg_max_kernel(const float* __restrict__ x,
                                                      const int*   __restrict__ batch,
                                                      float*       __restrict__ g,
                                                      int n_nodes) {
    const int t   = threadIdx.x;
    const int grp = t >> 6;          // sub-stream 0..3
    const int f4  = t & 63;          // float4 feature group 0..63

    long start = (long)blockIdx.x * 512 + (long)grp * 128;
    if (start >= n_nodes) return;
    long end = start + 128;
    if (end > n_nodes) end = n_nodes;

    float4 run = make_float4(NEG_INF, NEG_INF, NEG_INF, NEG_INF);
    int cur = batch[start];

    for (long i = start; i < end; ++i) {
        int bv = batch[i];
        if (bv != cur) {
            float* dst = g + (size_t)cur * DIM + f4 * 4;
            atomicMax(dst + 0, run.x);
            atomicMax(dst + 1, run.y);
            atomicMax(dst + 2, run.z);
            atomicMax(dst + 3, run.w);
            run = make_float4(NEG_INF, NEG_INF, NEG_INF, NEG_INF);
            cur = bv;
        }
        const float4* row = reinterpret_cast<const float4*>(x + (size_t)i * DIM);
        float4 v = row[f4];
        run.x = fmaxf(run.x, v.x);
        run.y = fmaxf(run.y, v.y);
        run.z = fmaxf(run.z, v.z);
        run.w = fmaxf(run.w, v.w);
    }
    float* dst = g + (size_t)cur * DIM + f4 * 4;
    atomicMax(dst + 0, run.x);
    atomicMax(dst + 1, run.y);
    atomicMax(dst + 2, run.z);
    atomicMax(dst + 3, run.w);
}

// ---------------------------------------------------------------------------
// Map untouched (-inf) entries (empty graphs) to 0, matching the reference's
// where(counts>0, seg, 0).
// ---------------------------------------------------------------------------
__global__ __launch_bounds__(256) void fix_neginf_kernel(float* __restrict__ g, int n) {
    int i = blockIdx.x * 256 + threadIdx.x;
    if (i < n) {
        float v = g[i];
        g[i] = (v == NEG_INF) ? 0.0f : v;
    }
}

// ---------------------------------------------------------------------------
// FP32 GEMM + bias (+ReLU) using V_WMMA_F32_16X16X4_F32.
// M = 512 rows. One wave computes one 16x16 tile of C; 8 waves per block.
// A: [M,K] row-major. B: [K,N] row-major. C: [M,N].
// ---------------------------------------------------------------------------
template <int K, int N, bool RELU>
__global__ __launch_bounds__(256) void gemm_wmma_f32_kernel(const float* __restrict__ A,
                                                            const float* __restrict__ B,
                                                            const float* __restrict__ bias,
                                                            float*       __restrict__ C) {
    const int lane = threadIdx.x & 31;
    const int wave = threadIdx.x >> 5;
    const int half = lane >> 4;      // 0: lanes 0-15, 1: lanes 16-31
    const int l16  = lane & 15;

    const int tiles_n = N / 16;
    const int waveId  = blockIdx.x * 8 + wave;
    const int m0 = (waveId / tiles_n) * 16;
    const int n0 = (waveId % tiles_n) * 16;

    v8f c = {};  // zero accumulator

    // A fragment: lane holds row (m0+l16), K-slots {2*half, 2*half+1} per step.
    // B fragment: lane holds col (n0+l16), same K-slot pattern.
    const float* arow = A + (size_t)(m0 + l16) * K;
    const float* bcol = B + n0 + l16;

    for (int k = 0; k < K; k += 4) {
        const int kk = k + 2 * half;
        v2f a, b;
        a.x = arow[kk];
        a.y = arow[kk + 1];
        b.x = bcol[(size_t)kk * N];
        b.y = bcol[(size_t)(kk + 1) * N];
        c = __builtin_amdgcn_wmma_f32_16x16x4_f32(
                /*neg_a=*/false, a, /*neg_b=*/false, b,
                /*c_mod=*/(short)0, c, /*reuse_a=*/false, /*reuse_b=*/false);
    }

    // C/D layout: VGPR j -> row m0 + 8*half + j, col n0 + l16
    const int col = n0 + l16;
    const float bc = bias[col];
    const int row_base = m0 + 8 * half;
#pragma unroll
    for (int j = 0; j < 8; ++j) {
        float v = c[j] + bc;
        if (RELU) v = fmaxf(v, 0.0f);
        C[(size_t)(row_base + j) * N + col] = v;
    }
}

// ---------------------------------------------------------------------------
// Final skinny layer: out[g,o] = h2[g,:] . W3[:,o] + b3[o]   (N=2)
// ---------------------------------------------------------------------------
__global__ __launch_bounds__(256) void out_layer_kernel(const float* __restrict__ h,
                                                        const float* __restrict__ W3,
                                                        const float* __restrict__ b3,
                                                        float*       __restrict__ out) {
    int idx = blockIdx.x * 256 + threadIdx.x;     // 0..1023
    if (idx >= NGRAPHS * 2) return;
    int gi = idx >> 1;
    int o  = idx & 1;
    const float* hr = h + (size_t)gi * HID;
    float s = b3[o];
    for (int k = 0; k < HID; ++k) s += hr[k] * W3[k * 2 + o];
    out[idx] = s;
}

// ---------------------------------------------------------------------------
extern "C" void kernel_launch(void* const* d_in, const int* in_sizes, int n_in,
                              void* d_out, int out_size, void* d_ws, size_t ws_size,
                              hipStream_t stream) {
    const float* x     = (const float*)d_in[0];
    const int*   batch = (const int*)  d_in[1];
    const float* W1    = (const float*)d_in[2];
    const float* b1    = (const float*)d_in[3];
    const float* W2    = (const float*)d_in[4];
    const float* b2    = (const float*)d_in[5];
    const float* W3    = (const float*)d_in[6];
    const float* b3    = (const float*)d_in[7];
    float*       out   = (float*)d_out;

    const int n_nodes = in_sizes[0] / DIM;

    // workspace layout: g [512x256] | h1 [512x512] | h2 [512x512]
    float* g  = (float*)d_ws;
    float* h1 = g  + NGRAPHS * DIM;
    float* h2 = h1 + NGRAPHS * HID;

    // 1) init pooling buffer to -inf
    fill_neginf_kernel<<<(NGRAPHS * DIM + 255) / 256, 256, 0, stream>>>(g, NGRAPHS * DIM);

    // 2) segment max (sorted batch, run-reduced, boundary atomics)
    int nblk = (n_nodes + 511) / 512;
    seg_max_kernel<<<nblk, 256, 0, stream>>>(x, batch, g, n_nodes);

    // 3) empty graphs -> 0
    fix_neginf_kernel<<<(NGRAPHS * DIM + 255) / 256, 256, 0, stream>>>(g, NGRAPHS * DIM);

    // 4) h1 = relu(g @ W1 + b1)   M=512, K=256, N=512 -> 1024 tiles / 8 waves = 128 blocks
    gemm_wmma_f32_kernel<DIM, HID, true><<<(NGRAPHS / 16) * (HID / 16) / 8, 256, 0, stream>>>(g, W1, b1, h1);

    // 5) h2 = relu(h1 @ W2 + b2)  M=512, K=512, N=512
    gemm_wmma_f32_kernel<HID, HID, true><<<(NGRAPHS / 16) * (HID / 16) / 8, 256, 0, stream>>>(h1, W2, b2, h2);

    // 6) out = h2 @ W3 + b3       [512,2]
    out_layer_kernel<<<(NGRAPHS * 2 + 255) / 256, 256, 0, stream>>>(h2, W3, b3, out);
}